// GraphNormGAT_33251636806237
// MI455X (gfx1250) — compile-verified
//
#include <hip/hip_runtime.h>

typedef float v2f __attribute__((ext_vector_type(2)));
typedef float v8f __attribute__((ext_vector_type(8)));

#define FEAT 128
#define EPSV 1e-5f

// One workgroup per graph. 8 waves; wave w reduces features [16w, 16w+16)
// over all nodes of the graph using V_WMMA_F32_16X16X4_F32 with A == ones:
//   D[m][n] = sum_k 1 * B[k][n] + C[m][n]   -> every row of D = column sums.
// B tile (4 nodes x 16 features) per lane (matches A-layout mirrored):
//   lanes 0-15 : VGPR0 = x[k0+0][f], VGPR1 = x[k0+1][f]
//   lanes 16-31: VGPR0 = x[k0+2][f], VGPR1 = x[k0+3][f]
__global__ __launch_bounds__(256)
void graphnorm_wmma_kernel(const float* __restrict__ x,
                           const float* __restrict__ gamma,
                           const float* __restrict__ beta,
                           const int* __restrict__ batch,
                           float* __restrict__ out) {
    __shared__ float s_sum[FEAT];
    __shared__ float s_sq[FEAT];
    __shared__ float s_scale[FEAT];
    __shared__ float s_shift[FEAT];

    const int g     = blockIdx.x;
    const int segs  = batch[g];
    const int sege  = batch[g + 1];
    const int count = sege - segs;

    const int t    = threadIdx.x;
    const int lane = t & 31;
    const int wave = t >> 5;
    const int half = lane >> 4;   // 0: K=0,1   1: K=2,3
    const int col  = lane & 15;
    const int f    = wave * 16 + col;   // feature owned in pass 1

    const float* __restrict__ xg = x   + (size_t)segs * FEAT;
    float* __restrict__       og = out + (size_t)segs * FEAT;

    // ---------------- Pass 1: sums via WMMA (matrix-pipe reduction) --------
    v8f accS = {0.f, 0.f, 0.f, 0.f, 0.f, 0.f, 0.f, 0.f};
    v8f accQ = {0.f, 0.f, 0.f, 0.f, 0.f, 0.f, 0.f, 0.f};
    v2f ones; ones.x = 1.0f; ones.y = 1.0f;

    const int fullChunks = count >> 2;   // uniform across the block
    for (int c = 0; c < fullChunks; ++c) {
        const int k0 = (c << 2) + (half << 1);
        const float v0 = xg[(size_t)k0 * FEAT + f];
        const float v1 = xg[(size_t)(k0 + 1) * FEAT + f];
        v2f b;  b.x  = v0;      b.y  = v1;
        v2f b2; b2.x = v0 * v0; b2.y = v1 * v1;
        accS = __builtin_amdgcn_wmma_f32_16x16x4_f32(
            false, ones, false, b,  (short)0, accS, false, false);
        accQ = __builtin_amdgcn_wmma_f32_16x16x4_f32(
            false, ones, false, b2, (short)0, accQ, false, false);
    }
    if (count & 3) {   // uniform condition: EXEC all-ones at the WMMAs
        const int k0 = (fullChunks << 2) + (half << 1);
        const int k1 = k0 + 1;
        const float v0 = (k0 < count) ? xg[(size_t)k0 * FEAT + f] : 0.0f;
        const float v1 = (k1 < count) ? xg[(size_t)k1 * FEAT + f] : 0.0f;
        v2f b;  b.x  = v0;      b.y  = v1;
        v2f b2; b2.x = v0 * v0; b2.y = v1 * v1;
        accS = __builtin_amdgcn_wmma_f32_16x16x4_f32(
            false, ones, false, b,  (short)0, accS, false, false);
        accQ = __builtin_amdgcn_wmma_f32_16x16x4_f32(
            false, ones, false, b2, (short)0, accQ, false, false);
    }

    // Row M=0 of the 16x16 f32 accumulator lives in VGPR0 of lanes 0-15.
    if (lane < 16) {
        s_sum[f] = accS[0];
        s_sq[f]  = accQ[0];
    }
    __syncthreads();

    // ---------------- Per-feature affine params ---------------------------
    if (t < FEAT) {
        const float cnt  = (float)count;
        const float mean = s_sum[t] / cnt;
        float var = s_sq[t] - cnt * mean * mean;        // unbiased (ddof=1)
        var = (count > 1) ? (var / (cnt - 1.0f)) : 0.0f;
        var = fmaxf(var, 0.0f);
        const float sc = gamma[t] / (sqrtf(var) + EPSV);
        s_scale[t] = sc;
        s_shift[t] = fmaf(-mean, sc, beta[t]);
    }
    __syncthreads();

    // ---------------- Pass 2: streamed normalize (b128 in / b128 out) -----
    // Threads 0-31 cover one full 512B row: fully coalesced.
    const int frow = (t & 31) << 2;   // feature quad
    const int n0   = t >> 5;          // node offset, 8 nodes per block-iter
    const float4 sc4 = *(const float4*)&s_scale[frow];
    const float4 sh4 = *(const float4*)&s_shift[frow];
    for (int n = n0; n < count; n += 8) {
        const float4 v = *(const float4*)(xg + (size_t)n * FEAT + frow);
        float4 o;
        o.x = fmaf(v.x, sc4.x, sh4.x);
        o.y = fmaf(v.y, sc4.y, sh4.y);
        o.z = fmaf(v.z, sc4.z, sh4.z);
        o.w = fmaf(v.w, sc4.w, sh4.w);
        *(float4*)(og + (size_t)n * FEAT + frow) = o;
    }
}

extern "C" void kernel_launch(void* const* d_in, const int* in_sizes, int n_in,
                              void* d_out, int out_size, void* d_ws, size_t ws_size,
                              hipStream_t stream) {
    const float* x     = (const float*)d_in[0];
    const float* gamma = (const float*)d_in[1];
    const float* beta  = (const float*)d_in[2];
    const int*   batch = (const int*)d_in[3];
    float*       out   = (float*)d_out;

    const int G = in_sizes[3] - 1;   // number of graphs
    // in_sizes[1] == FEAT == 128 per the reference shapes.

    graphnorm_wmma_kernel<<<dim3(G), dim3(256), 0, stream>>>(x, gamma, beta, batch, out);
}